// RP_68461778698458
// MI455X (gfx1250) — compile-verified
//
#include <hip/hip_runtime.h>

typedef __attribute__((ext_vector_type(2))) float v2f;
typedef __attribute__((ext_vector_type(8))) float v8f;

#define ROW_L 2048
#define S_OUT 65
#define THRESH 0.2f
#define EPS_V 1e-6f
#define WAVES_PER_BLOCK 8
#define VEC_PER_LANE 16  // 16 float4 * 32 lanes * 4 floats = 2048

__device__ __forceinline__ float wave_sum_f32(float v) {
#pragma unroll
  for (int off = 16; off > 0; off >>= 1) v += __shfl_xor(v, off, 32);
  return v;
}
__device__ __forceinline__ int wave_min_i32(int v) {
#pragma unroll
  for (int off = 16; off > 0; off >>= 1) v = min(v, __shfl_xor(v, off, 32));
  return v;
}
__device__ __forceinline__ int wave_max_i32(int v) {
#pragma unroll
  for (int off = 16; off > 0; off >>= 1) v = max(v, __shfl_xor(v, off, 32));
  return v;
}

__global__ __launch_bounds__(WAVES_PER_BLOCK * 32) void rp_wmma_kernel(
    const float* __restrict__ x, float* __restrict__ out, int nrows) {
  __shared__ float xs_s[WAVES_PER_BLOCK][80];

  const int lane = threadIdx.x & 31;
  const int wave = threadIdx.x >> 5;
  const int row = blockIdx.x * WAVES_PER_BLOCK + wave;
  const float* __restrict__ xr = x + (size_t)row * ROW_L;
  const float4* __restrict__ xr4 = (const float4*)xr;

  // ---- coalesced load of the whole row into registers (16 x b128 / lane) ----
  float4 v[VEC_PER_LANE];
#pragma unroll
  for (int k = 0; k < VEC_PER_LANE; ++k) v[k] = xr4[k * 32 + lane];

  // ---- pass 1: sum + nonzero span ----
  float s = 0.f;
  int firstnz = 1 << 30, lastnz = -1;
#pragma unroll
  for (int k = 0; k < VEC_PER_LANE; ++k) {
    const int base = (k * 32 + lane) * 4;
    const float e0 = v[k].x, e1 = v[k].y, e2 = v[k].z, e3 = v[k].w;
    s += e0; s += e1; s += e2; s += e3;
    if (e0 != 0.f) { firstnz = min(firstnz, base + 0); lastnz = max(lastnz, base + 0); }
    if (e1 != 0.f) { firstnz = min(firstnz, base + 1); lastnz = max(lastnz, base + 1); }
    if (e2 != 0.f) { firstnz = min(firstnz, base + 2); lastnz = max(lastnz, base + 2); }
    if (e3 != 0.f) { firstnz = min(firstnz, base + 3); lastnz = max(lastnz, base + 3); }
  }
  s = wave_sum_f32(s);
  const int start = wave_min_i32(firstnz);
  const int end = wave_max_i32(lastnz);
  const bool has_data = (end >= start);
  const float cnt = has_data ? (float)(end - start + 1) : 1.0f;
  const float mean = s / cnt;

  // ---- pass 2: masked variance (register resident, no reload) ----
  float ss = 0.f;
#pragma unroll
  for (int k = 0; k < VEC_PER_LANE; ++k) {
    const int base = (k * 32 + lane) * 4;
    const float e[4] = {v[k].x, v[k].y, v[k].z, v[k].w};
#pragma unroll
    for (int c = 0; c < 4; ++c) {
      const int idx = base + c;
      if (idx >= start && idx <= end) {
        const float t = e[c] - mean;
        ss += t * t;
      }
    }
  }
  const float var = wave_sum_f32(ss) / cnt;
  const float stdv = fmaxf(sqrtf(var), EPS_V);

  // ---- normalize first 65 elements into LDS (slab of 80; pad slots hold data or sentinels) ----
  // Invalid positions get a sentinel 1024+idx: valid normalized values are
  // bounded by sqrt(cnt) <= ~45.3 (since max|x-mean| <= sqrt(cnt*var) and
  // std >= sqrt(var)), so any pair touching an invalid index has
  // |diff| >= ~978 >> 0.2, and two invalid indices differ by >= 1.0.
  // This makes rp*mask_sq == (|diff| < 0.2), removing all mask logic from
  // the WMMA tile loop.
  // lane l (k==0) holds global indices 4l..4l+3; lanes 0..19 cover 0..79
  if (lane < 20) {
    const float e[4] = {v[0].x, v[0].y, v[0].z, v[0].w};
#pragma unroll
    for (int c = 0; c < 4; ++c) {
      const int idx = lane * 4 + c;
      float xn;
      if (has_data && idx >= start && idx <= end)
        xn = (e[c] - mean) / stdv;
      else
        xn = 1024.0f + (float)idx;  // sentinel: kills the pair in the compare
      xs_s[wave][idx] = xn;
    }
  }
  __syncthreads();

  // ---- per-lane K=0 fragment values for the 5 tiles (lanes 0-15 active) ----
  float xt[5];
#pragma unroll
  for (int t = 0; t < 5; ++t) {
    const float val = xs_s[wave][t * 16 + (lane & 15)];
    xt[t] = (lane < 16) ? val : 0.f;
  }
  const float onev = (lane < 16) ? 1.f : 0.f;
  const v2f ones = {onev, 0.f};
  const v2f negones = {-onev, 0.f};

  float* __restrict__ orow = out + (size_t)row * (S_OUT * S_OUT);
  const int Nc = lane & 15;
  const int Mbase = (lane >> 4) << 3;  // 0 or 8

#pragma unroll
  for (int ti = 0; ti < 5; ++ti) {
    const v2f a = {xt[ti], 0.f};
    v8f c0 = {};
    // d1[i][j] = xs_i * (-1)  (exact); depends only on ti -> hoisted
    const v8f d1 = __builtin_amdgcn_wmma_f32_16x16x4_f32(
        false, a, false, negones, (short)0, c0, false, false);
#pragma unroll
    for (int tj = 0; tj < 5; ++tj) {
      const v2f b = {xt[tj], 0.f};
      // d2[i][j] = 1 * xs_j + d1 = fl(xs_j - xs_i)  (exact FMA, mult by 1.0)
      const v8f d2 = __builtin_amdgcn_wmma_f32_16x16x4_f32(
          false, ones, false, b, (short)0, d1, false, false);

      const int j = tj * 16 + Nc;
#pragma unroll
      for (int r8 = 0; r8 < 8; ++r8) {
        const int i = ti * 16 + Mbase + r8;
        const float o = (__builtin_fabsf(d2[r8]) < THRESH) ? 1.f : 0.f;
        // bounds are compile-time-true for ti,tj <= 3 after full unroll
        if (i < S_OUT && j < S_OUT) orow[i * S_OUT + j] = o;
      }
    }
  }
  (void)nrows;
}

extern "C" void kernel_launch(void* const* d_in, const int* in_sizes, int n_in,
                              void* d_out, int out_size, void* d_ws,
                              size_t ws_size, hipStream_t stream) {
  const float* x = (const float*)d_in[0];
  float* out = (float*)d_out;
  const int nrows = in_sizes[0] / ROW_L;  // 1024 * 16 = 16384
  dim3 block(WAVES_PER_BLOCK * 32);
  dim3 grid((nrows + WAVES_PER_BLOCK - 1) / WAVES_PER_BLOCK);
  rp_wmma_kernel<<<grid, block, 0, stream>>>(x, out, nrows);
  (void)n_in; (void)out_size; (void)d_ws; (void)ws_size;
}